// GraphSAGEModel_6674379178178
// MI455X (gfx1250) — compile-verified
//
#include <hip/hip_runtime.h>

// GraphSAGE 2-layer, N=100000 nodes, E=1000000 edges, D=128, fp32.
// Edge phase: coalesced float4 gather + global_atomic_add_f32 scatter (L2-resident).
// Node phase: fused mean-scale + dual GEMM (W_l on mean, W_r on root) + bias (+ReLU)
// using V_WMMA_F32_16X16X4_F32 (exact fp32 tensor path on CDNA5).

#define DFEAT 128
#define NNODES 100000
#define NEDGES 1000000

typedef float v2f __attribute__((ext_vector_type(2)));
typedef float v8f __attribute__((ext_vector_type(8)));

// ---------------------------------------------------------------- zero fill
__global__ void zero_f32v4(float4* __restrict__ p, long n4) {
  long i = (long)blockIdx.x * blockDim.x + threadIdx.x;
  if (i < n4) p[i] = make_float4(0.f, 0.f, 0.f, 0.f);
}

// ---------------------------------------------------------------- degrees
__global__ void degree_k(const int* __restrict__ dst, float* __restrict__ cnt, int E) {
  int e = blockIdx.x * blockDim.x + threadIdx.x;
  if (e < E) atomicAdd(&cnt[dst[e]], 1.0f);
}

__global__ void invdeg_k(const float* __restrict__ cnt, float* __restrict__ invc, int n) {
  int i = blockIdx.x * blockDim.x + threadIdx.x;
  if (i < n) invc[i] = 1.0f / fmaxf(cnt[i], 1.0f);
}

// ---------------------------------------------------------------- edge scatter-add
// One thread per (edge, 4-feature chunk): 32 consecutive lanes cover one source row
// -> fully coalesced 512B gather per edge; scatter via 4x global_atomic_add_f32.
__global__ void scatter_k(const float* __restrict__ feat,
                          const int* __restrict__ src,
                          const int* __restrict__ dst,
                          float* __restrict__ agg, int E) {
  long gid = (long)blockIdx.x * blockDim.x + threadIdx.x;
  int e = (int)(gid >> 5);
  if (e >= E) return;
  int c = (int)(gid & 31);
  int s = src[e];
  int d = dst[e];
  const float4 v = *(const float4*)(feat + (long)s * DFEAT + c * 4);
  float* o = agg + (long)d * DFEAT + c * 4;
  atomicAdd(o + 0, v.x);
  atomicAdd(o + 1, v.y);
  atomicAdd(o + 2, v.z);
  atomicAdd(o + 3, v.w);
}

// ---------------------------------------------------------------- fused SAGE node update
// out[n,:] = act( (agg[n,:]*invc[n]) @ Wl + bl + root[n,:] @ Wr )
// Block = 256 thr = 8 waves; block owns a 16-row panel, wave w owns cols [16w,16w+16).
// K-loop: 32 steps of V_WMMA_F32_16X16X4_F32, two terms per step.
__global__ __launch_bounds__(256) void sage_gemm_k(
    const float* __restrict__ agg, const float* __restrict__ invc,
    const float* __restrict__ root,
    const float* __restrict__ Wl, const float* __restrict__ bl,
    const float* __restrict__ Wr,
    float* __restrict__ out, int do_relu) {
  const int lane = threadIdx.x & 31;
  const int wave = threadIdx.x >> 5;
  const int rowBase = blockIdx.x * 16;
  const int colBase = wave * 16;
  const int m = lane & 15;          // M index carried by this lane (A); N index for B and C
  const int half = lane >> 4;       // lane half selects K sub-pair / M+8
  const int koff = half << 1;       // 0 or 2
  const int col = colBase + m;      // N index carried by this lane (B and C)

  const int mrow = rowBase + m;
  const float inv = invc[mrow];
  const float2* aggp = (const float2*)(agg + (long)mrow * DFEAT);
  const float2* xp   = (const float2*)(root + (long)mrow * DFEAT);

  v8f c = {};
#pragma unroll 8
  for (int kb = 0; kb < DFEAT / 4; ++kb) {
    const int k0 = kb * 4 + koff;                 // even -> float2 aligned
    const float2 am = aggp[k0 >> 1];
    const float2 ax = xp[k0 >> 1];
    v2f a_l; a_l.x = am.x * inv; a_l.y = am.y * inv;   // mean on the fly
    v2f a_r; a_r.x = ax.x;       a_r.y = ax.y;
    v2f b_l; b_l.x = Wl[(long)k0 * DFEAT + col]; b_l.y = Wl[(long)(k0 + 1) * DFEAT + col];
    v2f b_r; b_r.x = Wr[(long)k0 * DFEAT + col]; b_r.y = Wr[(long)(k0 + 1) * DFEAT + col];
    c = __builtin_amdgcn_wmma_f32_16x16x4_f32(false, a_l, false, b_l, (short)0, c, false, false);
    c = __builtin_amdgcn_wmma_f32_16x16x4_f32(false, a_r, false, b_r, (short)0, c, false, false);
  }

  const float bias = bl[col];
#pragma unroll
  for (int r = 0; r < 8; ++r) {
    float v = c[r] + bias;
    if (do_relu) v = fmaxf(v, 0.0f);
    const int row = rowBase + r + (half << 3);    // lanes 16-31 hold M = r+8
    out[(long)row * DFEAT + col] = v;
  }
}

// ---------------------------------------------------------------- launch
extern "C" void kernel_launch(void* const* d_in, const int* in_sizes, int n_in,
                              void* d_out, int out_size, void* d_ws, size_t ws_size,
                              hipStream_t stream) {
  (void)in_sizes; (void)n_in; (void)out_size; (void)ws_size;
  const float* x   = (const float*)d_in[0];
  const int*   ei  = (const int*)d_in[1];
  const int*   src = ei;
  const int*   dst = ei + NEDGES;
  const float* W1l = (const float*)d_in[2];
  const float* b1  = (const float*)d_in[3];
  const float* W1r = (const float*)d_in[4];
  const float* W2l = (const float*)d_in[5];
  const float* b2  = (const float*)d_in[6];
  const float* W2r = (const float*)d_in[7];
  float* out = (float*)d_out;

  float* agg  = (float*)d_ws;                    // N*D
  float* cnt  = agg + (long)NNODES * DFEAT;      // N
  float* invc = cnt + NNODES;                    // N

  const int T = 256;
  const long aggN = (long)NNODES * DFEAT;

  // 1) zero agg + cnt (contiguous, N*D + N divisible by 4)
  {
    long n4 = (aggN + NNODES) / 4;
    zero_f32v4<<<dim3((unsigned)((n4 + T - 1) / T)), dim3(T), 0, stream>>>((float4*)agg, n4);
  }
  // 2) degrees + reciprocals (shared by both layers)
  degree_k<<<dim3((NEDGES + T - 1) / T), dim3(T), 0, stream>>>(dst, cnt, NEDGES);
  invdeg_k<<<dim3((NNODES + T - 1) / T), dim3(T), 0, stream>>>(cnt, invc, NNODES);

  const long eThreads = (long)NEDGES * 32;
  const unsigned eBlocks = (unsigned)((eThreads + T - 1) / T);
  const unsigned gBlocks = NNODES / 16;          // 6250, exact -> EXEC all ones in WMMA

  // 3) layer 1: scatter x, fused WMMA update (+ReLU) into d_out (acts as h)
  scatter_k<<<dim3(eBlocks), dim3(T), 0, stream>>>(x, src, dst, agg, NEDGES);
  sage_gemm_k<<<dim3(gBlocks), dim3(T), 0, stream>>>(agg, invc, x, W1l, b1, W1r, out, 1);

  // 4) layer 2: re-zero agg, scatter h, fused WMMA update (no ReLU) in-place on d_out
  {
    long n4 = aggN / 4;
    zero_f32v4<<<dim3((unsigned)((n4 + T - 1) / T)), dim3(T), 0, stream>>>((float4*)agg, n4);
  }
  scatter_k<<<dim3(eBlocks), dim3(T), 0, stream>>>(out, src, dst, agg, NEDGES);
  sage_gemm_k<<<dim3(gBlocks), dim3(T), 0, stream>>>(agg, invc, out, W2l, b2, W2r, out, 0);
}